// TopKSAE_38422777430237
// MI455X (gfx1250) — compile-verified
//
#include <hip/hip_runtime.h>
#include <stddef.h>
#include <stdint.h>

// Problem constants (from reference setup_inputs)
#define BROWS 4096
#define NIN   768
#define NLAT  24576
#define KSEL  32
#define AUXK  512

typedef float v2f __attribute__((ext_vector_type(2)));
typedef float v8f __attribute__((ext_vector_type(8)));
typedef unsigned int u32;
typedef u32 v4u __attribute__((ext_vector_type(4)));
typedef int  v8i __attribute__((ext_vector_type(8)));
typedef int  v4i __attribute__((ext_vector_type(4)));

#if __has_builtin(__builtin_amdgcn_tensor_load_to_lds) && \
    __has_builtin(__builtin_amdgcn_s_wait_tensorcnt)
#define HAVE_TDM 1
#else
#define HAVE_TDM 0
#endif

// ---------------------------------------------------------------------------
// Kernel 1: transpose W_dec [NIN, NLAT] -> WdT [NLAT, NIN]
// (decoder columns become contiguous rows; W_dec ~75MB, fully L2-resident)
// ---------------------------------------------------------------------------
__global__ __launch_bounds__(256) void transpose_wdec(
    const float* __restrict__ Wdec, float* __restrict__ WdT) {
  __shared__ float tile[32][33];
  const int l0 = blockIdx.x * 32;   // latent dim
  const int i0 = blockIdx.y * 32;   // input dim
  const int tx = threadIdx.x;       // 0..31
  for (int r = threadIdx.y; r < 32; r += 8)
    tile[r][tx] = Wdec[(size_t)(i0 + r) * NLAT + (l0 + tx)];
  __syncthreads();
  for (int r = threadIdx.y; r < 32; r += 8)
    WdT[(size_t)(l0 + r) * NIN + (i0 + tx)] = tile[tx][r];
}

// ---------------------------------------------------------------------------
// Kernel 2a: fold the input centering into a per-latent bias:
//   c[n] = b_enc[n] - sum_k b[k] * W_enc[n,k]
// so the GEMM streams raw x tiles:  alpha_pre = x @ W_enc^T + c
// ---------------------------------------------------------------------------
__global__ __launch_bounds__(256) void enc_bias(
    const float* __restrict__ bvec, const float* __restrict__ Wenc,
    const float* __restrict__ benc, float* __restrict__ cvec) {
  const int n = blockIdx.x * blockDim.x + threadIdx.x;  // 0..NLAT-1
  const float* wr = Wenc + (size_t)n * NIN;
  float acc = 0.f;
  for (int k = 0; k < NIN; k += 4) {
    float4 w  = *(const float4*)(wr + k);
    float4 bb = *(const float4*)(bvec + k);
    acc += w.x * bb.x + w.y * bb.y + w.z * bb.z + w.w * bb.w;
  }
  cvec[n] = benc[n] - acc;
}

// ---------------------------------------------------------------------------
// TDM helper: async-DMA one 128-row x 32-col fp32 tile (row stride NIN) from
// global into LDS, padding every 32 DWORDs with 4 DWORDs so the LDS image has
// a 36-float row stride (bank-conflict padding done by the Tensor Data Mover).
// D# layout per CDNA5 ISA ch.8 (group0: count/lds/global/type, group1: dims).
// This toolchain's builtin takes the 6-arg form:
//   (v4u g0, v8i g1, v4i g2, v4i g3, v8i extra, i32 cpol)
// groups 2/3 (+extra) are zero: tile_dim2/3 = 0 => unused (2D tile).
// ---------------------------------------------------------------------------
#if HAVE_TDM
__device__ __forceinline__ void tdm_load_tile(const void* gsrc, u32 ldsAddr,
                                              u32 rows /* tensor dim1 */) {
  const unsigned long long ga = (unsigned long long)(uintptr_t)gsrc;
  v4u g0;
  g0[0] = 1u;                                   // count=1 (valid), user mode
  g0[1] = ldsAddr;                              // LDS byte address
  g0[2] = (u32)(ga & 0xFFFFFFFFu);              // global_addr[31:0]
  g0[3] = (u32)((ga >> 32) & 0x01FFFFFFu)       // global_addr[56:32]
        | (2u << 30);                           // type = 2 ("image")
  const u32 d0 = NIN;                           // tensor dim0 (elements)
  v8i g1;
  g1[0] = (int)((2u << 16)                      // data_size = 4 bytes
              | (1u << 20)                      // pad_enable
              | (4u << 22)                      // pad_interval: 32 DWORDs
              | (3u << 25));                    // pad_amount:   4 DWORDs
  g1[1] = (int)((d0 & 0xFFFFu) << 16);          // abar_addr=0 | dim0 lo16
  g1[2] = (int)(((d0 >> 16) & 0xFFFFu) | ((rows & 0xFFFFu) << 16));
  g1[3] = (int)(((rows >> 16) & 0xFFFFu) | (32u << 16));  // tile_dim0 = 32
  g1[4] = (int)128;                             // tile_dim1=128, tile_dim2=0
  g1[5] = (int)NIN;                             // tensor_dim0_stride lo32
  g1[6] = 0;                                    // stride hi | dim1_stride lo
  g1[7] = 0;
  v4i z4 = {0, 0, 0, 0};
  v8i z8 = {0, 0, 0, 0, 0, 0, 0, 0};
  __builtin_amdgcn_tensor_load_to_lds(g0, g1, z4, z4, z8, 0);
}
#endif

// ---------------------------------------------------------------------------
// Kernel 2b: encoder GEMM via V_WMMA_F32_16X16X4_F32.
//   alpha_pre[m,n] = sum_k x[m,k]*W_enc[n,k] + c[n]
// Block: 256 thr (8 waves), tile 128x128, K-chunk 32, double-buffered LDS fed
// by the TDM (wave 0 issues, s_wait_tensorcnt + barrier synchronize).
// Each wave: 16 rows x 128 cols = 8 WMMA accumulator tiles (64 acc VGPRs).
// ---------------------------------------------------------------------------
#define KC  32
#define LDK 36
#define NCHUNK (NIN / KC)  // 24

__global__ __launch_bounds__(256) void enc_gemm(
    const float* __restrict__ x, const float* __restrict__ Wenc,
    const float* __restrict__ cvec, float* __restrict__ alpha_pre) {
  __shared__ float sA[2][128 * LDK];
  __shared__ float sB[2][128 * LDK];

  const int tid   = threadIdx.x;
  const int wave  = tid >> 5;
  const int lane  = tid & 31;
  const int laneM = lane & 15;
  const int laneH = lane >> 4;
  const int gm0 = blockIdx.y * 128;
  const int gn0 = blockIdx.x * 128;

  const float* Abase = x    + (size_t)gm0 * NIN;  // tile row origin in x
  const float* Bbase = Wenc + (size_t)gn0 * NIN;  // tile row origin in W_enc

  v8f acc[8];
#pragma unroll
  for (int i = 0; i < 8; ++i) acc[i] = (v8f){0.f,0.f,0.f,0.f,0.f,0.f,0.f,0.f};

#if HAVE_TDM
  if (tid < 32) {  // wave 0 drives the Tensor Data Mover
    tdm_load_tile(Abase, (u32)(uintptr_t)&sA[0][0], BROWS);
    tdm_load_tile(Bbase, (u32)(uintptr_t)&sB[0][0], NLAT);
  }
#endif

  for (int kc = 0; kc < NCHUNK; ++kc) {
    const int cur = kc & 1;
#if HAVE_TDM
    if (tid < 32) {
      if (kc + 1 < NCHUNK) {  // prefetch next chunk into the other buffer
        tdm_load_tile(Abase + (kc + 1) * KC, (u32)(uintptr_t)&sA[cur ^ 1][0],
                      BROWS);
        tdm_load_tile(Bbase + (kc + 1) * KC, (u32)(uintptr_t)&sB[cur ^ 1][0],
                      NLAT);
        __builtin_amdgcn_s_wait_tensorcnt((short)2);  // chunk kc landed
      } else {
        __builtin_amdgcn_s_wait_tensorcnt((short)0);
      }
    }
#else
    // Fallback: cooperative float4 staging (256 thr * 4 passes = 128x32 tile)
    {
      const int lr = tid >> 3;
      const int lc = (tid & 7) * 4;
#pragma unroll
      for (int p = 0; p < 4; ++p) {
        const int r = lr + p * 32;
        *(float4*)&sA[cur][r * LDK + lc] =
            *(const float4*)(Abase + (size_t)r * NIN + kc * KC + lc);
        *(float4*)&sB[cur][r * LDK + lc] =
            *(const float4*)(Bbase + (size_t)r * NIN + kc * KC + lc);
      }
    }
#endif
    __syncthreads();  // tile kc visible to all waves

#pragma unroll
    for (int kk = 0; kk < KC; kk += 4) {
      // A 16x4 fragment (ISA: lanes 0-15 hold K{0,1}, lanes 16-31 K{2,3})
      v2f a;
      a.x = sA[cur][(wave * 16 + laneM) * LDK + kk + laneH * 2 + 0];
      a.y = sA[cur][(wave * 16 + laneM) * LDK + kk + laneH * 2 + 1];
#pragma unroll
      for (int nt = 0; nt < 8; ++nt) {
        // B 4x16 fragment: K rows striped across lane halves per VGPR
        v2f bf;
        bf.x = sB[cur][(nt * 16 + laneM) * LDK + kk + laneH];
        bf.y = sB[cur][(nt * 16 + laneM) * LDK + kk + 2 + laneH];
        acc[nt] = __builtin_amdgcn_wmma_f32_16x16x4_f32(
            false, a, false, bf, (short)0, acc[nt], false, false);
      }
    }
    __syncthreads();  // compute done before buffer is overwritten
  }

  // Store C tiles (layout: VGPR v -> rows v / v+8 split across lane halves)
#pragma unroll
  for (int nt = 0; nt < 8; ++nt) {
    const int gcol = gn0 + nt * 16 + laneM;
    const float cv = cvec[gcol];
#pragma unroll
    for (int v = 0; v < 8; ++v) {
      const int grow = gm0 + wave * 16 + v + laneH * 8;
      alpha_pre[(size_t)grow * NLAT + gcol] = acc[nt][v] + cv;
    }
  }
}

// ---------------------------------------------------------------------------
// Kernel 3: per-row dual top-k (k=32 dense; k=512 over dead-masked values)
// One 1024-thread block per row; whole row (96KB) lives in LDS as order-
// preserving uint keys. 4-pass radix select finds the exact k-th threshold;
// a deterministic block-scan compaction emits (idx,val) lists + alpha + mask.
// ---------------------------------------------------------------------------
__device__ __forceinline__ unsigned f2key(float f) {
  unsigned u = __float_as_uint(f);
  return (u & 0x80000000u) ? ~u : (u | 0x80000000u);
}
__device__ __forceinline__ float key2f(unsigned key) {
  unsigned u = (key & 0x80000000u) ? (key & 0x7FFFFFFFu) : ~key;
  return __uint_as_float(u);
}
__device__ __forceinline__ unsigned effkey(const unsigned* keys,
                                           const unsigned* deadw,
                                           bool useMask, int i) {
  unsigned k = keys[i];
  if (useMask && !((deadw[i >> 5] >> (i & 31)) & 1u)) k = 0u;  // below -inf
  return k;
}

__device__ void radix_select(const unsigned* keys, const unsigned* deadw,
                             bool useMask, int k, int tid, int bdim, int* hist,
                             unsigned& T, int& needEq, unsigned* shPrefix,
                             int* shRemaining) {
  if (tid == 0) { *shPrefix = 0u; *shRemaining = k; }
  __syncthreads();
  for (int byte = 3; byte >= 0; --byte) {
    const int shift = byte * 8;
    for (int b = tid; b < 256; b += bdim) hist[b] = 0;
    __syncthreads();
    const unsigned prefix = *shPrefix;
    const unsigned maskHi = (byte == 3) ? 0u : (0xFFFFFFFFu << ((byte + 1) * 8));
    for (int i = tid; i < NLAT; i += bdim) {
      const unsigned kk = effkey(keys, deadw, useMask, i);
      if ((kk & maskHi) == (prefix & maskHi))
        atomicAdd(&hist[(kk >> shift) & 255], 1);
    }
    __syncthreads();
    if (tid == 0) {
      const int rem = *shRemaining;
      int cum = 0, sel = 0;
      for (int b = 255; b >= 0; --b) {
        const int c = hist[b];
        if (cum + c >= rem) { sel = b; break; }
        cum += c;
      }
      *shPrefix = prefix | ((unsigned)sel << shift);
      *shRemaining = rem - cum;  // still needed among keys == prefix
    }
    __syncthreads();
  }
  T = *shPrefix;
  needEq = *shRemaining;
  __syncthreads();
}

__device__ void select_write(const unsigned* keys, const unsigned* deadw,
                             bool useMask, unsigned T, int needEq, int tid,
                             int bdim, int* sGT, int* sEQ, int* shTotalGT,
                             float* alphaOut, float* firedOut, int* idxOut,
                             float* valOut) {
  int cg = 0, ce = 0;
  for (int i = tid; i < NLAT; i += bdim) {
    const unsigned kk = effkey(keys, deadw, useMask, i);
    cg += (kk > T) ? 1 : 0;
    ce += (kk == T) ? 1 : 0;
  }
  sGT[tid] = cg; sEQ[tid] = ce;
  __syncthreads();
  if (tid == 0) {  // serial exclusive scan: 2*bdim adds, negligible
    int ag = 0, ae = 0;
    for (int t = 0; t < bdim; ++t) {
      const int g = sGT[t]; sGT[t] = ag; ag += g;
      const int e = sEQ[t]; sEQ[t] = ae; ae += e;
    }
    *shTotalGT = ag;
  }
  __syncthreads();
  int gpos = sGT[tid];
  int epos = sEQ[tid];
  const int totalGT = *shTotalGT;
  for (int i = tid; i < NLAT; i += bdim) {
    const unsigned kk = effkey(keys, deadw, useMask, i);
    bool sel = false;
    int pos = 0;
    if (kk > T) {
      sel = true; pos = gpos++;
    } else if (kk == T) {
      const int er = epos++;              // deterministic tie rank
      if (er < needEq) { sel = true; pos = totalGT + er; }
    }
    const float val = key2f(keys[i]);     // original (unmasked) value
    if (alphaOut) {
      alphaOut[i] = sel ? val : 0.0f;
      firedOut[i] = sel ? 1.0f : 0.0f;
    }
    if (sel) { idxOut[pos] = i; valOut[pos] = val; }
  }
  __syncthreads();
}

__global__ __launch_bounds__(1024) void topk_kernel(
    float* __restrict__ alpha /* in: alpha_pre, out: sparse alpha */,
    float* __restrict__ fired, const int* __restrict__ miss_counts,
    int* __restrict__ mainIdx, float* __restrict__ mainVal,
    int* __restrict__ auxIdx, float* __restrict__ auxVal) {
  extern __shared__ unsigned keys[];        // NLAT keys = 96KB dynamic LDS
  __shared__ unsigned deadw[NLAT / 32];     // dead-feature bitmask (3KB)
  __shared__ int hist[256];
  __shared__ int sGT[1024];
  __shared__ int sEQ[1024];
  __shared__ unsigned shPrefix;
  __shared__ int shRemaining;
  __shared__ int shTotalGT;

  const int tid = threadIdx.x, bdim = blockDim.x;
  const int row = blockIdx.x;
  float* arow = alpha + (size_t)row * NLAT;
  float* frow = fired + (size_t)row * NLAT;

  for (int i = tid; i < NLAT; i += bdim) keys[i] = f2key(arow[i]);
  for (int w = tid; w < NLAT / 32; w += bdim) {
    unsigned m = 0;
    for (int b = 0; b < 32; ++b)
      m |= (miss_counts[w * 32 + b] >= 1 ? 1u : 0u) << b;
    deadw[w] = m;
  }
  __syncthreads();

  unsigned T; int needEq;
  // main top-32 (also rewrites alpha + fired_mask in place)
  radix_select(keys, deadw, false, KSEL, tid, bdim, hist, T, needEq,
               &shPrefix, &shRemaining);
  select_write(keys, deadw, false, T, needEq, tid, bdim, sGT, sEQ, &shTotalGT,
               arow, frow, mainIdx + (size_t)row * KSEL,
               mainVal + (size_t)row * KSEL);
  // aux top-512 among dead features
  radix_select(keys, deadw, true, AUXK, tid, bdim, hist, T, needEq,
               &shPrefix, &shRemaining);
  select_write(keys, deadw, true, T, needEq, tid, bdim, sGT, sEQ, &shTotalGT,
               nullptr, nullptr, auxIdx + (size_t)row * AUXK,
               auxVal + (size_t)row * AUXK);
}

// ---------------------------------------------------------------------------
// Kernel 4: sparse decode  out[row,:] = b + sum_j val_j * WdT[idx_j, :]
// WdT rows are contiguous (768 floats) and L2-resident.
// ---------------------------------------------------------------------------
__global__ __launch_bounds__(256) void decode_kernel(
    const int* __restrict__ idxList, const float* __restrict__ valList,
    int count, const float* __restrict__ WdT, const float* __restrict__ bvec,
    float* __restrict__ out) {
  __shared__ int sIdx[AUXK];
  __shared__ float sVal[AUXK];
  const int row = blockIdx.x;
  const int* il = idxList + (size_t)row * count;
  const float* vl = valList + (size_t)row * count;
  for (int j = threadIdx.x; j < count; j += blockDim.x) {
    sIdx[j] = il[j];
    sVal[j] = vl[j];
  }
  __syncthreads();
  for (int col = threadIdx.x; col < NIN; col += blockDim.x) {
    float acc = bvec[col];
    for (int j = 0; j < count; ++j)
      acc = fmaf(sVal[j], WdT[(size_t)sIdx[j] * NIN + col], acc);
    out[(size_t)row * NIN + col] = acc;
  }
}

// ---------------------------------------------------------------------------
// Launch
// ---------------------------------------------------------------------------
extern "C" void kernel_launch(void* const* d_in, const int* in_sizes, int n_in,
                              void* d_out, int out_size, void* d_ws,
                              size_t ws_size, hipStream_t stream) {
  const float* x    = (const float*)d_in[0];
  const float* bvec = (const float*)d_in[1];
  const float* Wenc = (const float*)d_in[2];
  const float* benc = (const float*)d_in[3];
  const float* Wdec = (const float*)d_in[4];
  const int*   miss = (const int*)d_in[5];
  // d_in[6]=k(32), d_in[7]=aux_k(512): fixed per reference setup.

  float* out     = (float*)d_out;
  float* x_hat   = out;                              // [B, NIN]
  float* alpha   = x_hat + (size_t)BROWS * NIN;      // [B, NLAT] (GEMM scratch too)
  float* fired   = alpha + (size_t)BROWS * NLAT;     // [B, NLAT]
  float* aux_hat = fired + (size_t)BROWS * NLAT;     // [B, NIN]

  char* ws = (char*)d_ws;
  int*   mIdx = (int*)ws;   ws += (size_t)BROWS * KSEL * sizeof(int);
  float* mVal = (float*)ws; ws += (size_t)BROWS * KSEL * sizeof(float);
  int*   aIdx = (int*)ws;   ws += (size_t)BROWS * AUXK * sizeof(int);
  float* aVal = (float*)ws; ws += (size_t)BROWS * AUXK * sizeof(float);
  float* cvec = (float*)ws; ws += (size_t)NLAT * sizeof(float);
  float* WdT  = (float*)ws;  // [NLAT, NIN] ~75.5MB

  transpose_wdec<<<dim3(NLAT / 32, NIN / 32), dim3(32, 8), 0, stream>>>(Wdec, WdT);
  enc_bias<<<NLAT / 256, 256, 0, stream>>>(bvec, Wenc, benc, cvec);
  enc_gemm<<<dim3(NLAT / 128, BROWS / 128), 256, 0, stream>>>(x, Wenc, cvec,
                                                              alpha);
  topk_kernel<<<BROWS, 1024, NLAT * sizeof(unsigned), stream>>>(
      alpha, fired, miss, mIdx, mVal, aIdx, aVal);
  decode_kernel<<<BROWS, 256, 0, stream>>>(mIdx, mVal, KSEL, WdT, bvec, x_hat);
  decode_kernel<<<BROWS, 256, 0, stream>>>(aIdx, aVal, AUXK, WdT, bvec, aux_hat);
}